// GraphMerge_33062658244939
// MI455X (gfx1250) — compile-verified
//
#include <hip/hip_runtime.h>
#include <math.h>

// ---------------------------------------------------------------------------
// GraphMerge (3-hop GNN) for gfx1250 / MI455X.
//   per hop:  Hn = SpMM(A, H);  H' = Hn @ Wn^T + bn + H @ Ws^T + bs; ELU(not last)
// SpMM: wave-per-edge float4 gather + global_atomic_add_f32 scatter (L2-resident).
// Dense: fused fp32 WMMA GEMM, K=512 ([Hn|H] @ [Wn^T;Ws^T]); A/B tiles staged
//        into double-buffered LDS by the Tensor Data Mover (tensor_load_to_lds,
//        s_wait_tensorcnt), overlapped with v_wmma_f32_16x16x4_f32 compute.
// ---------------------------------------------------------------------------

typedef float v2f __attribute__((ext_vector_type(2)));
typedef float v8f __attribute__((ext_vector_type(8)));
typedef unsigned int u32x4 __attribute__((ext_vector_type(4)));
typedef int i32x4 __attribute__((ext_vector_type(4)));
typedef int i32x8 __attribute__((ext_vector_type(8)));

#define DIM 256
#define BM 64            // M rows per block
#define KC 32            // K chunk staged in LDS
#define KTOT (2 * DIM)   // concatenated K = 512
#define LSTRIDE 36       // LDS row stride (floats): 32 data + 4 pad (bank spread, 16B aligned)

#if __has_builtin(__builtin_amdgcn_tensor_load_to_lds) && \
    __has_builtin(__builtin_amdgcn_s_wait_tensorcnt)
#define USE_TDM 1
#else
#define USE_TDM 0
#endif

// --------------------------- SpMM (scatter-add) ----------------------------
__global__ __launch_bounds__(256) void spmm_kernel(
    const float* __restrict__ H, const float* __restrict__ vals,
    const int* __restrict__ rows, const int* __restrict__ cols,
    float* __restrict__ accum, int nnz)
{
    int edge = blockIdx.x * 8 + (threadIdx.x >> 5);
    if (edge >= nnz) return;
    int lane = threadIdx.x & 31;
    int r = rows[edge];
    int c = cols[edge];
    float v = vals[edge];

    const float4* src = (const float4*)(H + (size_t)c * DIM);
    float*        dst = accum + (size_t)r * DIM;

    float4 x0 = src[lane];
    float4 x1 = src[lane + 32];

    int o = lane * 4;
    atomicAdd(dst + o + 0, x0.x * v);
    atomicAdd(dst + o + 1, x0.y * v);
    atomicAdd(dst + o + 2, x0.z * v);
    atomicAdd(dst + o + 3, x0.w * v);
    int o2 = o + 128;
    atomicAdd(dst + o2 + 0, x1.x * v);
    atomicAdd(dst + o2 + 1, x1.y * v);
    atomicAdd(dst + o2 + 2, x1.z * v);
    atomicAdd(dst + o2 + 3, x1.w * v);
}

// ----------------------------- TDM helpers ---------------------------------
__device__ __forceinline__ unsigned ldsOffset(const void* p) {
    // LDS aperture: low 32 bits of the flat address are the LDS byte offset
    return (unsigned)(uintptr_t)p;
}

#if USE_TDM
// 2D tile load: tileRows x 32 f32 elements, source rows of rowStrideElems
// floats, into LDS at ldsOff with a row stride of LSTRIDE floats (pad 4 dw
// every 32 dw). Rows >= rowsAvail are zero-filled by the TDM bounds check.
__device__ __forceinline__ void tdm_load_tile(
    unsigned ldsOff, const float* gsrc, unsigned rowsAvail, unsigned tileRows,
    unsigned rowStrideElems)
{
    unsigned long long ga = (unsigned long long)(uintptr_t)gsrc;
    u32x4 g0;
    g0.x = 1u;                                            // count=1, user mode
    g0.y = ldsOff;                                        // lds_addr
    g0.z = (unsigned)(ga & 0xFFFFFFFFu);                  // global_addr[31:0]
    g0.w = (unsigned)((ga >> 32) & 0x01FFFFFFu) | (2u << 30); // addr[56:32]|type=2

    i32x8 g1;
    // data_size=4B (2), pad_enable, pad_interval=32 dwords (4), pad_amount=4 dw (3)
    g1[0] = (int)((2u << 16) | (1u << 20) | (4u << 22) | (3u << 25));
    g1[1] = (int)(256u << 16);                 // tensor_dim0[15:0]=256 (row len)
    g1[2] = (int)(rowsAvail << 16);            // dim0 hi=0 | tensor_dim1 lo
    g1[3] = (int)((rowsAvail >> 16) | (32u << 16)); // dim1 hi | tile_dim0=32
    g1[4] = (int)(tileRows & 0xFFFFu);         // tile_dim1 | tile_dim2=0
    g1[5] = (int)rowStrideElems;               // tensor_dim0_stride[31:0]
    g1[6] = 0;                                 // stride hi | dim1_stride lo
    g1[7] = 0;                                 // dim1_stride hi
    i32x4 z4 = {0, 0, 0, 0};                   // D# groups 2/3 unused (2D tensor)
    i32x8 z8 = {0, 0, 0, 0, 0, 0, 0, 0};
    __builtin_amdgcn_tensor_load_to_lds(g0, g1, z4, z4, z8, 0);
}
#endif

// ------------------- fused dense GEMM + bias + ELU (WMMA) ------------------
// out[m,n] = sum_k [Hn|H][m,k] * [Wn^T;Ws^T][k,n] + bn[n] + bs[n]
__global__ __launch_bounds__(256) void gemm_fused_kernel(
    const float* __restrict__ Hn, const float* __restrict__ H,
    const float* __restrict__ Wn, const float* __restrict__ Ws,
    const float* __restrict__ bn, const float* __restrict__ bs,
    float* __restrict__ out, int N, int applyElu)
{
    __shared__ float As[2][BM * LSTRIDE];    // 2 x 9 KB   : A rows [m][k]
    __shared__ float Bs[2][DIM * LSTRIDE];   // 2 x 36 KB  : B as [n][k] (= W rows)

    int tid    = threadIdx.x;
    int lane   = tid & 31;
    int waveId = tid >> 5;       // 8 waves; wave owns N range [waveId*32, +32)
    int lcol   = lane & 15;
    int half   = lane >> 4;      // 0: lanes 0-15, 1: lanes 16-31
    int tileM  = blockIdx.x * BM;
    int n0     = waveId * 32;

    v8f acc[4][2];
#pragma unroll
    for (int ms = 0; ms < 4; ++ms)
#pragma unroll
        for (int ns = 0; ns < 2; ++ns)
            acc[ms][ns] = (v8f){0.f, 0.f, 0.f, 0.f, 0.f, 0.f, 0.f, 0.f};

#if USE_TDM
    // ---- issue chunk 0 (A + B tiles) via the Tensor Data Mover
    if (waveId == 0) {
        const float* aSrc = Hn + (size_t)tileM * DIM;        // kc==0 -> Hn, col 0
        tdm_load_tile(ldsOffset(&As[0][0]), aSrc,
                      (unsigned)(N - tileM), BM, DIM);
        tdm_load_tile(ldsOffset(&Bs[0][0]), Wn,              // kc==0 -> Wn, col 0
                      DIM, DIM, DIM);
        __builtin_amdgcn_s_wait_tensorcnt(0);
    }
    __syncthreads();
#endif

    int buf = 0;
    for (int kc = 0; kc < KTOT; kc += KC) {
#if USE_TDM
        int nkc = kc + KC;
        if (waveId == 0 && nkc < KTOT) {
            // async-prefetch next K chunk into the other buffer (overlaps WMMA)
            int kcol = nkc & (DIM - 1);
            const float* aSrc = ((nkc < DIM) ? Hn : H) + (size_t)tileM * DIM + kcol;
            const float* bSrc = ((nkc < DIM) ? Wn : Ws) + kcol;
            tdm_load_tile(ldsOffset(&As[buf ^ 1][0]), aSrc,
                          (unsigned)(N - tileM), BM, DIM);
            tdm_load_tile(ldsOffset(&Bs[buf ^ 1][0]), bSrc, DIM, DIM, DIM);
        }
#else
        {   // manual staging fallback: contiguous float4 tile copies
            int kcol = kc & (DIM - 1);
            const float* aSrc = ((kc < DIM) ? Hn : H);
            const float* bSrc = ((kc < DIM) ? Wn : Ws);
            for (int idx = tid; idx < BM * 8; idx += 256) {      // 64 rows x 8 f4
                int rr = idx >> 3, c4 = idx & 7;
                int grow = tileM + rr; if (grow >= N) grow = N - 1;
                float4 vq = *(const float4*)(aSrc + (size_t)grow * DIM + kcol + c4 * 4);
                *(float4*)(&As[buf][rr * LSTRIDE + c4 * 4]) = vq;
            }
            for (int idx = tid; idx < DIM * 8; idx += 256) {     // 256 rows x 8 f4
                int n = idx >> 3, c4 = idx & 7;
                float4 vq = *(const float4*)(bSrc + (size_t)n * DIM + kcol + c4 * 4);
                *(float4*)(&Bs[buf][n * LSTRIDE + c4 * 4]) = vq;
            }
            __syncthreads();
        }
#endif

#pragma unroll
        for (int ks = 0; ks < KC; ks += 4) {
            int kA = ks + 2 * half;    // lanes 0-15 -> K0/K1, 16-31 -> K2/K3
            v2f a[4], b[2];
#pragma unroll
            for (int ms = 0; ms < 4; ++ms) {
                int base = (ms * 16 + lcol) * LSTRIDE + kA;
                v2f av; av.x = As[buf][base]; av.y = As[buf][base + 1];
                a[ms] = av;
            }
#pragma unroll
            for (int ns = 0; ns < 2; ++ns) {
                int base = (n0 + ns * 16 + lcol) * LSTRIDE + kA;
                v2f bv; bv.x = Bs[buf][base]; bv.y = Bs[buf][base + 1];
                b[ns] = bv;
            }
#pragma unroll
            for (int ms = 0; ms < 4; ++ms)
#pragma unroll
                for (int ns = 0; ns < 2; ++ns)
                    acc[ms][ns] = __builtin_amdgcn_wmma_f32_16x16x4_f32(
                        false, a[ms], false, b[ns], (short)0,
                        acc[ms][ns], false, false);
        }

#if USE_TDM
        if (waveId == 0) __builtin_amdgcn_s_wait_tensorcnt(0);
        __syncthreads();   // next buffer ready + everyone done with current
#else
        __syncthreads();
#endif
        buf ^= 1;
    }

    // ---- epilogue: bias + optional ELU. C layout: VGPR v -> rows v, v+8
#pragma unroll
    for (int ns = 0; ns < 2; ++ns) {
        int col = n0 + ns * 16 + lcol;
        float bt = bn[col] + bs[col];
#pragma unroll
        for (int ms = 0; ms < 4; ++ms) {
#pragma unroll
            for (int v = 0; v < 8; ++v) {
                int row = tileM + ms * 16 + half * 8 + v;
                if (row < N) {
                    float x = acc[ms][ns][v] + bt;
                    if (applyElu) x = (x > 0.f) ? x : (expf(x) - 1.0f);
                    out[(size_t)row * DIM + col] = x;
                }
            }
        }
    }
}

// ------------------------------- launcher ----------------------------------
extern "C" void kernel_launch(void* const* d_in, const int* in_sizes, int n_in,
                              void* d_out, int out_size, void* d_ws, size_t ws_size,
                              hipStream_t stream) {
    const float* m    = (const float*)d_in[0];
    const float* vals = (const float*)d_in[1];
    const float* Wn   = (const float*)d_in[2];
    const float* bn   = (const float*)d_in[3];
    const float* Ws   = (const float*)d_in[4];
    const float* bs   = (const float*)d_in[5];
    const int*   rows = (const int*)d_in[6];
    const int*   cols = (const int*)d_in[7];

    int N    = in_sizes[0] / DIM;
    int nnz  = in_sizes[1];
    int hops = in_sizes[2] / (DIM * DIM);

    float* accum = (float*)d_ws;   // N*DIM f32 SpMM accumulator
    float* out   = (float*)d_out;

    int gemmBlocks = (N + BM - 1) / BM;
    int spmmBlocks = (nnz + 7) / 8;

    for (int i = 0; i < hops; ++i) {
        const float* Hcur = (i == 0) ? m : out;  // ping through d_out in place
        (void)hipMemsetAsync(accum, 0, (size_t)N * DIM * sizeof(float), stream);
        spmm_kernel<<<spmmBlocks, 256, 0, stream>>>(Hcur, vals, rows, cols,
                                                    accum, nnz);
        gemm_fused_kernel<<<gemmBlocks, 256, 0, stream>>>(
            accum, Hcur,
            Wn + (size_t)i * DIM * DIM, Ws + (size_t)i * DIM * DIM,
            bn + (size_t)i * DIM, bs + (size_t)i * DIM,
            out, N, (i != hops - 1) ? 1 : 0);
    }
}